// CircularNN_65283502899762
// MI455X (gfx1250) — compile-verified
//
#include <hip/hip_runtime.h>

// Problem constants (from reference)
#define D        784
#define NCLS     10
#define ROWS     16      // batch rows per workgroup == WMMA M
#define STRIDE   800     // padded row stride in LDS (floats)
#define NTHREADS 256     // 8 wave32s
#define NWAVES   8

typedef __attribute__((ext_vector_type(2))) float v2f;
typedef __attribute__((ext_vector_type(8))) float v8f;

__device__ __forceinline__ float gelu_exact(float x) {
    // jax.nn.gelu(approximate=False): 0.5*x*(1+erf(x/sqrt(2)))
    return 0.5f * x * (1.0f + erff(x * 0.70710678118654752f));
}

// One sparse "circular" layer entirely in LDS: out[n] = gelu(b[n] + sum_k in[idx[n,k]]*w[n,k])
template<int K>
__device__ __forceinline__ void sparse_layer_lds(
    const float* __restrict__ in,   // [ROWS][STRIDE] in LDS
    float*       __restrict__ out,  // [ROWS][STRIDE] in LDS
    const int*   __restrict__ idx,  // [D][K] global (L2-resident)
    const float* __restrict__ w,    // [D][K]
    const float* __restrict__ b,    // [D]
    int tid)
{
    for (int n = tid; n < D; n += NTHREADS) {
        int   ii[K];
        float ww[K];
#pragma unroll
        for (int k = 0; k < K; ++k) { ii[k] = idx[n * K + k]; ww[k] = w[n * K + k]; }
        const float bb = b[n];
#pragma unroll
        for (int r = 0; r < ROWS; ++r) {
            float acc = bb;
#pragma unroll
            for (int k = 0; k < K; ++k)
                acc = fmaf(in[r * STRIDE + ii[k]], ww[k], acc);
            out[r * STRIDE + n] = gelu_exact(acc);
        }
    }
}

__global__ __launch_bounds__(NTHREADS) void circular_nn_kernel(
    const float* __restrict__ x,
    const int*   __restrict__ idx1, const float* __restrict__ w1, const float* __restrict__ b1,
    const int*   __restrict__ idx2, const float* __restrict__ w2, const float* __restrict__ b2,
    const int*   __restrict__ idx3, const float* __restrict__ w3, const float* __restrict__ b3,
    const float* __restrict__ fc_w, const float* __restrict__ fc_b,
    float*       __restrict__ out)
{
    extern __shared__ float smem[];
    float* bufA = smem;                     // ROWS*STRIDE (x, h2; later: 8 partial logit tiles)
    float* bufB = smem + ROWS * STRIDE;     // ROWS*STRIDE (h1, h3)
    float* lgt  = smem + 2 * ROWS * STRIDE; // 16*16 reduced logits

    const int tid  = threadIdx.x;
    const int row0 = blockIdx.x * ROWS;

    // Stage 16 rows of x into LDS as float4 (784*4 and 800*4 are both /16 -> aligned)
    for (int i = tid; i < ROWS * (D / 4); i += NTHREADS) {
        const int r = i / (D / 4);
        const int c = i - r * (D / 4);
        const float4 v = ((const float4*)(x + (size_t)(row0 + r) * D))[c];
        *(float4*)&bufA[r * STRIDE + c * 4] = v;
    }
    __syncthreads();

    sparse_layer_lds<2>(bufA, bufB, idx1, w1, b1, tid);  // x  -> h1
    __syncthreads();
    sparse_layer_lds<4>(bufB, bufA, idx2, w2, b2, tid);  // h1 -> h2
    __syncthreads();
    sparse_layer_lds<8>(bufA, bufB, idx3, w3, b3, tid);  // h2 -> h3 (in bufB)
    __syncthreads();

    // FC: logits(16x16) = h3(16x784) @ fc_w^T(784x16, cols >= NCLS zeroed)
    // via V_WMMA_F32_16X16X4_F32. All 8 waves participate: wave w takes the
    // K-slice {w*4 + 32*i}, so each wave runs a short (~25) serial WMMA chain.
    {
        const int wave = tid >> 5;
        const int lane = tid & 31;
        const int m    = lane & 15;   // A: row M; B: col N
        const int half = lane >> 4;   // 0 -> K{0,1}, 1 -> K{2,3} per VGPR pair
        const bool nvalid = (m < NCLS);
        // Safe row pointer: invalid lanes read row 0 (in-bounds), value zeroed below.
        const float* fw = fc_w + (nvalid ? m : 0) * D;
        const float  sel = nvalid ? 1.0f : 0.0f;

        v8f c = {};
        for (int k0 = wave * 4; k0 < D; k0 += NWAVES * 4) {
            const int kk = k0 + 2 * half;
            // A[m][kk..kk+1] from LDS: one ds_load_b64 (8B aligned: STRIDE, kk even)
            const v2f a = *(const v2f*)&bufB[m * STRIDE + kk];
            // B[kk..kk+1][m] = fc_w[m][kk..kk+1]: one global_load_b64, branch-free
            const float2 bv = *(const float2*)&fw[kk];
            v2f bm;
            bm.x = bv.x * sel;
            bm.y = bv.y * sel;
            c = __builtin_amdgcn_wmma_f32_16x16x4_f32(
                    /*neg_a=*/false, a, /*neg_b=*/false, bm,
                    /*c_mod=*/(short)0, c, /*reuse_a=*/false, /*reuse_b=*/false);
        }
        // Partial D tile -> LDS (bufA is dead after layer 3; reuse 8*256 floats of it).
        // D layout: VGPR v -> M = v (lanes 0-15) / v+8 (lanes 16-31), N = lane&15.
#pragma unroll
        for (int v = 0; v < 8; ++v)
            bufA[wave * 256 + (v + 8 * half) * 16 + m] = c[v];
    }
    __syncthreads();

    // Reduce the 8 per-wave partial tiles: thread tid owns logits element tid = row*16+col.
    {
        float s = 0.0f;
#pragma unroll
        for (int w = 0; w < NWAVES; ++w)
            s += bufA[w * 256 + tid];
        lgt[tid] = s;
    }
    __syncthreads();

    // Per-row softmax over the 10 valid logits
    if (tid < ROWS) {
        float l[NCLS];
        float mx = -3.402823466e38f;
#pragma unroll
        for (int cc = 0; cc < NCLS; ++cc) {
            l[cc] = lgt[tid * 16 + cc] + fc_b[cc];
            mx = fmaxf(mx, l[cc]);
        }
        float s = 0.0f;
#pragma unroll
        for (int cc = 0; cc < NCLS; ++cc) { l[cc] = __expf(l[cc] - mx); s += l[cc]; }
        const float inv = 1.0f / s;
        float* o = out + (size_t)(row0 + tid) * NCLS;
#pragma unroll
        for (int cc = 0; cc < NCLS; ++cc) o[cc] = l[cc] * inv;
    }
}

extern "C" void kernel_launch(void* const* d_in, const int* in_sizes, int n_in,
                              void* d_out, int out_size, void* d_ws, size_t ws_size,
                              hipStream_t stream) {
    (void)n_in; (void)out_size; (void)d_ws; (void)ws_size;

    const float* x    = (const float*)d_in[0];
    const int*   idx1 = (const int*)  d_in[1];
    const float* w1   = (const float*)d_in[2];
    const float* b1   = (const float*)d_in[3];
    const int*   idx2 = (const int*)  d_in[4];
    const float* w2   = (const float*)d_in[5];
    const float* b2   = (const float*)d_in[6];
    const int*   idx3 = (const int*)  d_in[7];
    const float* w3   = (const float*)d_in[8];
    const float* b3   = (const float*)d_in[9];
    const float* fcw  = (const float*)d_in[10];
    const float* fcb  = (const float*)d_in[11];
    float*       out  = (float*)d_out;

    const int Btot = in_sizes[0] / D;                       // 65536
    const size_t smem = (size_t)(2 * ROWS * STRIDE + 16 * 16) * sizeof(float); // ~103 KB

    // Allow >default dynamic LDS (CDNA5 WGP has 320 KB).
    (void)hipFuncSetAttribute((const void*)circular_nn_kernel,
                              hipFuncAttributeMaxDynamicSharedMemorySize, (int)smem);

    circular_nn_kernel<<<Btot / ROWS, NTHREADS, smem, stream>>>(
        x, idx1, w1, b1, idx2, w2, b2, idx3, w3, b3, fcw, fcb, out);
}